// get_loss_74586402062752
// MI455X (gfx1250) — compile-verified
//
#include <hip/hip_runtime.h>
#include <hip/hip_bf16.h>

// Chamfer loss on MI455X (gfx1250, wave32):
//   dist[m][n] = |p1_m|^2 + |p2_n|^2 - 2*p1_m.p2_n
// Cross terms via V_WMMA_F32_16X16X4_F32 (points padded to K=4, w=0; p1
// pre-scaled by -2).  The norm sums are folded into the WMMA C operand so the
// matrix pipe emits finished squared distances.  All 25 B operands + norms are
// preloaded into registers per wave, so the inner loop is WMMA + mins only
// (no LDS traffic, no dscnt waits).  Column mins are kept in registers and
// flushed once per wave with ds_min_num_f32 LDS float atomics.

typedef float v2f __attribute__((ext_vector_type(2)));
typedef float v8f __attribute__((ext_vector_type(8)));

#define NUM_T 28
#define NUM_P 400
#define NUM_B 32
#define NTILE (NUM_P / 16)   // 25 sixteen-point tiles per axis
#define FBIG  3.0e38f

__global__ __launch_bounds__(256) void chamfer_wmma_kernel(
    const float* __restrict__ pc_new,   // [32, 28*400, 3]
    const float* __restrict__ pc_gt,    // [32, 28*400, 3]
    float* __restrict__ out)            // [1]
{
    __shared__ float p1[NUM_P * 4];     // (-2x,-2y,-2z,0)
    __shared__ float p2[NUM_P * 4];     // ( x,  y,  z, 0)
    __shared__ float n1[NUM_P];
    __shared__ float n2[NUM_P];
    __shared__ float colmin[NUM_P];
    __shared__ float wg_acc;

    const int tid   = threadIdx.x;
    const int lane  = tid & 31;
    const int wave  = tid >> 5;
    const int nwav  = blockDim.x >> 5;
    const int half  = lane >> 4;        // 0: K/M lanes 0-15, 1: lanes 16-31
    const int l16   = lane & 15;

    const int bid = blockIdx.x;         // 0..895
    const int b   = bid / NUM_T;
    const int t   = bid % NUM_T;
    const size_t base = ((size_t)b * (NUM_T * NUM_P) + (size_t)t * NUM_P) * 3;

    // ---- stage points + norms into LDS -------------------------------------
    for (int p = tid; p < NUM_P; p += blockDim.x) {
        float x = pc_new[base + (size_t)p * 3 + 0];
        float y = pc_new[base + (size_t)p * 3 + 1];
        float z = pc_new[base + (size_t)p * 3 + 2];
        p1[p * 4 + 0] = -2.0f * x;
        p1[p * 4 + 1] = -2.0f * y;
        p1[p * 4 + 2] = -2.0f * z;
        p1[p * 4 + 3] = 0.0f;
        n1[p] = x * x + y * y + z * z;

        float gx = pc_gt[base + (size_t)p * 3 + 0];
        float gy = pc_gt[base + (size_t)p * 3 + 1];
        float gz = pc_gt[base + (size_t)p * 3 + 2];
        p2[p * 4 + 0] = gx;
        p2[p * 4 + 1] = gy;
        p2[p * 4 + 2] = gz;
        p2[p * 4 + 3] = 0.0f;
        n2[p] = gx * gx + gy * gy + gz * gz;

        colmin[p] = FBIG;
    }
    if (tid == 0) wg_acc = 0.0f;
    __syncthreads();

    // ---- preload all B operands + gt norms into registers (strip-invariant)
    v2f   bb[NTILE];
    float cn[NTILE];
    float vcol[NTILE];
#pragma unroll
    for (int j = 0; j < NTILE; ++j) {
        bb[j]   = *(const v2f*)&p2[(j * 16 + l16) * 4 + half * 2];
        cn[j]   = n2[j * 16 + l16];
        vcol[j] = FBIG;
    }

    // ---- per-wave row strips over the 400x400 distance matrix --------------
    float row_sum = 0.0f;   // sum of finished row-mins (valid on l16==0 lanes)

    for (int s = wave; s < NTILE; s += nwav) {
        // A operand: row m = s*16 + l16, components {half*2, half*2+1}
        v2f a = *(const v2f*)&p1[(s * 16 + l16) * 4 + half * 2];

        float rn[8];
#pragma unroll
        for (int r = 0; r < 8; ++r)
            rn[r] = n1[s * 16 + r + half * 8];   // M = r + 8*half (C/D layout)

        float rmin[8];
#pragma unroll
        for (int r = 0; r < 8; ++r) rmin[r] = FBIG;

#pragma unroll
        for (int j = 0; j < NTILE; ++j) {
            // Fold the norm sums into the WMMA accumulator:
            //   C[m][n] = |p1_m|^2 + |p2_n|^2  ->  D = C - 2*p1.p2 = dist
            v8f c;
#pragma unroll
            for (int r = 0; r < 8; ++r) c[r] = rn[r] + cn[j];

            c = __builtin_amdgcn_wmma_f32_16x16x4_f32(
                    /*neg_a=*/false, a, /*neg_b=*/false, bb[j],
                    /*c_mod=*/(short)0, c, /*reuse_a=*/false, /*reuse_b=*/false);

            float cpart = FBIG;
#pragma unroll
            for (int r = 0; r < 8; ++r) {
                rmin[r] = fminf(rmin[r], c[r]);
                cpart   = fminf(cpart, c[r]);
            }
            vcol[j] = fminf(vcol[j], cpart);
        }

        // finish row mins: butterfly min across the 16-lane half
#pragma unroll
        for (int r = 0; r < 8; ++r) {
            float v = rmin[r];
            v = fminf(v, __shfl_xor(v, 1, 16));
            v = fminf(v, __shfl_xor(v, 2, 16));
            v = fminf(v, __shfl_xor(v, 4, 16));
            v = fminf(v, __shfl_xor(v, 8, 16));
            if (l16 == 0) row_sum += v;          // unique strip per wave: no atomic
        }
    }

    // flush this wave's column partial mins (one DS float atomic per tile col)
#pragma unroll
    for (int j = 0; j < NTILE; ++j) {
        unsigned caddr = (unsigned)(size_t)&colmin[j * 16 + l16];
        asm volatile("ds_min_num_f32 %0, %1"
                     :: "v"(caddr), "v"(vcol[j]) : "memory");
    }

    if (l16 == 0) atomicAdd(&wg_acc, row_sum);

    // make the hand-rolled ds atomics visible before reading colmin
    asm volatile("s_wait_dscnt 0" ::: "memory");
    __syncthreads();

    // ---- sum column mins ----------------------------------------------------
    float csum = 0.0f;
    for (int p = tid; p < NUM_P; p += blockDim.x) csum += colmin[p];
    atomicAdd(&wg_acc, csum);
    __syncthreads();

    if (tid == 0) {
        const float scale = 1.0f / ((float)NUM_P * (float)NUM_B * (float)NUM_T);
        atomicAdd(out, wg_acc * scale);
    }
}

__global__ void zero_out_kernel(float* o) { o[0] = 0.0f; }

extern "C" void kernel_launch(void* const* d_in, const int* in_sizes, int n_in,
                              void* d_out, int out_size, void* d_ws, size_t ws_size,
                              hipStream_t stream) {
    const float* pc_new = (const float*)d_in[0];
    const float* pc_gt  = (const float*)d_in[1];
    float* out = (float*)d_out;

    zero_out_kernel<<<1, 1, 0, stream>>>(out);
    chamfer_wmma_kernel<<<NUM_B * NUM_T, 256, 0, stream>>>(pc_new, pc_gt, out);
}